// SkilledLoRA_13451837571847
// MI455X (gfx1250) — compile-verified
//
#include <hip/hip_runtime.h>
#include <hip/hip_bf16.h>

// Problem constants (from reference)
#define BS      8
#define SEQ     2048
#define IN_F    2048
#define OUT_F   2048
#define NSPLIT  4
#define NSKILL  8
#define RANK    16
#define DSPLIT  (IN_F / NSPLIT)    // 512
#define OSPLIT  (OUT_F / NSPLIT)   // 512
#define SCALING 2.0f               // 32 / 16

typedef float v2f __attribute__((ext_vector_type(2)));
typedef float v8f __attribute__((ext_vector_type(8)));

// ---------------------------------------------------------------------------
// Phase 1: mix per-skill LoRA factors.
//   A_mix[b][i][r]  = sum_s w[b][q][s] * lora_a[q][s][d][r]    (i = q*512+d)
//   B_mix[b][r][o]  = SCALING * sum_s w[b][q][s] * lora_b[q][s][r][d]  (o = q*512+d)
// One thread per output element; 524288 threads total, exact grid.
// ---------------------------------------------------------------------------
__global__ __launch_bounds__(256) void lora_mix_kernel(
    const float* __restrict__ weights,   // [BS][NSPLIT][NSKILL]
    const float* __restrict__ lora_a,    // [NSPLIT][NSKILL][DSPLIT][RANK]
    const float* __restrict__ lora_b,    // [NSPLIT][NSKILL][RANK][OSPLIT]
    float* __restrict__ A_mix,           // [BS][IN_F][RANK]
    float* __restrict__ B_mix)           // [BS][RANK][OUT_F]
{
    const int idx  = blockIdx.x * 256 + threadIdx.x;
    const int half = BS * IN_F * RANK;          // 262144

    if (idx < half) {
        const int b   = idx / (IN_F * RANK);
        const int rem = idx - b * (IN_F * RANK);
        const int i   = rem / RANK;
        const int r   = rem - i * RANK;
        const int q   = i / DSPLIT;
        const int d   = i - q * DSPLIT;
        const float* w  = weights + (b * NSPLIT + q) * NSKILL;
        const float* la = lora_a + ((size_t)(q * NSKILL) * DSPLIT + d) * RANK + r;
        float acc = 0.f;
        #pragma unroll
        for (int s = 0; s < NSKILL; ++s)
            acc = __builtin_fmaf(w[s], la[(size_t)s * DSPLIT * RANK], acc);
        A_mix[idx] = acc;
    } else {
        const int j   = idx - half;
        const int b   = j / (RANK * OUT_F);
        const int rem = j - b * (RANK * OUT_F);
        const int r   = rem / OUT_F;
        const int o   = rem - r * OUT_F;
        const int q   = o / OSPLIT;
        const int d   = o - q * OSPLIT;
        const float* w  = weights + (b * NSPLIT + q) * NSKILL;
        const float* lb = lora_b + ((size_t)(q * NSKILL) * RANK + r) * OSPLIT + d;
        float acc = 0.f;
        #pragma unroll
        for (int s = 0; s < NSKILL; ++s)
            acc = __builtin_fmaf(w[s], lb[(size_t)s * RANK * OSPLIT], acc);
        B_mix[j] = acc * SCALING;
    }
}

// ---------------------------------------------------------------------------
// Phase 2: fused low-rank matmul, one wave per 16-row seq tile.
//   T   = x[b, m0:m0+16, :] @ A_mix[b]          (16x16, K=2048, f32 WMMA)
//   out = T @ B_mix[b]                          (16x2048, K=16, f32 WMMA)
//
// WMMA f32 16x16x4 operand layouts (wave32):
//   A (16x4):  lane L holds A[M = L%16][k0 + 2*(L/16) + v], v = VGPR 0..1
//   B (4x16):  lane L holds B[k0 + 2*(L/16) + v][N = L%16]
//   C/D:       lane L, VGPR v holds D[M = v + 8*(L/16)][N = L%16]
// ---------------------------------------------------------------------------
__global__ __launch_bounds__(128) void skilled_lora_kernel(
    const float* __restrict__ x,       // [BS][SEQ][IN_F]
    const float* __restrict__ A_mix,   // [BS][IN_F][RANK]
    const float* __restrict__ B_mix,   // [BS][RANK][OUT_F]  (pre-scaled)
    float* __restrict__ out)           // [BS][SEQ][OUT_F]
{
    __shared__ float Tlds[4][16][16];  // per-wave 16x16 transpose staging

    const int wave = threadIdx.x >> 5;
    const int lane = threadIdx.x & 31;
    const int h    = lane >> 4;        // lane half (0/1)
    const int c    = lane & 15;        // column / row-in-half index

    const int tile = blockIdx.x * 4 + wave;        // 0 .. 1023
    const int b    = tile >> 7;                    // / 128
    const int m0   = (tile & 127) << 4;            // * 16

    // ---------------- GEMM1: T = x_tile @ A_mix ----------------
    const float* xrow = x + ((size_t)b * SEQ + m0 + c) * IN_F + 2 * h;
    const float* acol = A_mix + (size_t)b * IN_F * RANK + (2 * h) * RANK + c;

    v8f acc = {0.f, 0.f, 0.f, 0.f, 0.f, 0.f, 0.f, 0.f};

    for (int k0 = 0; k0 < IN_F; k0 += 64) {
        if (k0 + 64 < IN_F)
            __builtin_prefetch(xrow + k0 + 64, 0, 0);  // global_prefetch_b8
        #pragma unroll
        for (int k = k0; k < k0 + 64; k += 4) {
            v2f av = *(const v2f*)(xrow + k);          // x[m0+c][k+2h .. +1]
            v2f bv;
            bv.x = acol[(size_t)k * RANK];             // A_mix[b][k+2h  ][c]
            bv.y = acol[(size_t)k * RANK + RANK];      // A_mix[b][k+2h+1][c]
            acc = __builtin_amdgcn_wmma_f32_16x16x4_f32(
                false, av, false, bv, (short)0, acc, false, false);
        }
    }

    // ---------------- transpose T via LDS (C/D layout -> A layout) ----------
    #pragma unroll
    for (int v = 0; v < 8; ++v)
        Tlds[wave][v + 8 * h][c] = acc[v];

    asm volatile("s_wait_dscnt 0x0" ::: "memory");

    v2f a2[4];
    #pragma unroll
    for (int kk = 0; kk < 4; ++kk)
        a2[kk] = *(const v2f*)&Tlds[wave][c][4 * kk + 2 * h];  // T[M=c][k0+2h ..]

    // ---------------- GEMM2: out_tile = T @ B_mix ----------------
    const float* Bbase = B_mix + (size_t)b * RANK * OUT_F + (2 * h) * OUT_F + c;
    float*       obase = out + ((size_t)b * SEQ + m0 + 8 * h) * OUT_F + c;

    for (int n0 = 0; n0 < OUT_F; n0 += 16) {
        v8f acc2 = {0.f, 0.f, 0.f, 0.f, 0.f, 0.f, 0.f, 0.f};
        const float* bp = Bbase + n0;
        #pragma unroll
        for (int kk = 0; kk < 4; ++kk) {
            v2f bv;
            bv.x = bp[(size_t)(4 * kk)     * OUT_F];   // B_mix[b][k0+2h  ][n0+c]
            bv.y = bp[(size_t)(4 * kk + 1) * OUT_F];   // B_mix[b][k0+2h+1][n0+c]
            acc2 = __builtin_amdgcn_wmma_f32_16x16x4_f32(
                false, a2[kk], false, bv, (short)0, acc2, false, false);
        }
        float* op = obase + n0;
        #pragma unroll
        for (int v = 0; v < 8; ++v)
            op[(size_t)v * OUT_F] = acc2[v];           // out[m0+v+8h][n0+c]
    }
}

// ---------------------------------------------------------------------------
extern "C" void kernel_launch(void* const* d_in, const int* in_sizes, int n_in,
                              void* d_out, int out_size, void* d_ws, size_t ws_size,
                              hipStream_t stream) {
    const float* input   = (const float*)d_in[0];  // [8][2048][2048]
    const float* weights = (const float*)d_in[1];  // [8][4][8]
    const float* lora_a  = (const float*)d_in[2];  // [4][8][512][16]
    const float* lora_b  = (const float*)d_in[3];  // [4][8][16][512]
    float*       outp    = (float*)d_out;          // [8][2048][2048]

    float* A_mix = (float*)d_ws;                               // 1 MB
    float* B_mix = A_mix + (size_t)BS * IN_F * RANK;           // 1 MB

    // Phase 1: 524288 elements, 256 threads/block -> 2048 blocks (exact)
    lora_mix_kernel<<<2048, 256, 0, stream>>>(weights, lora_a, lora_b, A_mix, B_mix);

    // Phase 2: 1024 tiles (8 batches x 128 seq tiles), 4 waves/block -> 256 blocks
    skilled_lora_kernel<<<256, 128, 0, stream>>>(input, A_mix, B_mix, outp);
}